// ContextWeaver_73426760892610
// MI455X (gfx1250) — compile-verified
//
#include <hip/hip_runtime.h>

typedef __attribute__((ext_vector_type(2))) float v2f;
typedef __attribute__((ext_vector_type(8))) float v8f;

#define NN   8192   // num nodes (M and N of the GEMM)
#define KTOT 64     // K*D = 2*32, GEMM reduction dim

// Each wave computes a 64x32 output block as 4x2 tiles of 16x16 using
// V_WMMA_F32_16X16X4_F32 (exact fp32 semantics, matching the reference).
// Block = 256 threads = 8 waves arranged 2 (i) x 4 (j) -> 128x128 per block.
__global__ __launch_bounds__(256)
void ContextWeaver_wmma_f32(const float* __restrict__ node,
                            const float* __restrict__ edge,
                            float* __restrict__ out) {
    const int lane   = threadIdx.x & 31;
    const int waveId = threadIdx.x >> 5;
    const int lane16 = lane & 15;
    const int half   = lane >> 4;      // 0: K pair {k,k+1}, 1: {k+2,k+3}

    const int wi = waveId & 1;         // wave position in i (0..1)
    const int wj = waveId >> 1;        // wave position in j (0..3)

    const int ibase = blockIdx.y * 128 + wi * 64;   // 4 M-tiles of 16
    const int jbase = blockIdx.x * 128 + wj * 32;   // 2 N-tiles of 16

    v8f acc[4][2] = {};   // 8 accumulators * 8 VGPRs = 64 VGPRs

    // Per-lane base pointers: A lane holds row (ibase+t*16+lane16),
    // K-offset 2*half within each k-step of 4 (ISA 7.12.2 32-bit A layout).
    const float* an[4];
#pragma unroll
    for (int t = 0; t < 4; ++t)
        an[t] = node + (size_t)(ibase + t * 16 + lane16) * KTOT + 2 * half;
    const float* bn[2];
#pragma unroll
    for (int t = 0; t < 2; ++t)
        bn[t] = edge + (size_t)(jbase + t * 16 + lane16) * KTOT + 2 * half;

#pragma unroll
    for (int kk = 0; kk < KTOT; kk += 4) {
        v2f a[4], b[2];
#pragma unroll
        for (int t = 0; t < 4; ++t) a[t] = *(const v2f*)(an[t] + kk);   // 8B coalesced
#pragma unroll
        for (int t = 0; t < 2; ++t) b[t] = *(const v2f*)(bn[t] + kk);
#pragma unroll
        for (int mi = 0; mi < 4; ++mi)
#pragma unroll
            for (int nj = 0; nj < 2; ++nj)
                acc[mi][nj] = __builtin_amdgcn_wmma_f32_16x16x4_f32(
                    /*neg_a=*/false, a[mi], /*neg_b=*/false, b[nj],
                    /*c_mod=*/(short)0, acc[mi][nj],
                    /*reuse_a=*/false, /*reuse_b=*/false);
    }

    // Store: C/D layout -> VGPR r: lanes 0-15 = row (base+r), lanes 16-31 = row (base+r+8).
    // ReLU + branchless diagonal zero; non-temporal (output is 268MB write-once > L2).
#pragma unroll
    for (int mi = 0; mi < 4; ++mi) {
#pragma unroll
        for (int nj = 0; nj < 2; ++nj) {
            const int col = jbase + nj * 16 + lane16;
#pragma unroll
            for (int r = 0; r < 8; ++r) {
                const int row = ibase + mi * 16 + r + 8 * half;
                float v = acc[mi][nj][r];
                v = fmaxf(v, 0.0f);
                v = (row == col) ? 0.0f : v;
                __builtin_nontemporal_store(v, out + (size_t)row * NN + col);
            }
        }
    }
}

extern "C" void kernel_launch(void* const* d_in, const int* in_sizes, int n_in,
                              void* d_out, int out_size, void* d_ws, size_t ws_size,
                              hipStream_t stream) {
    const float* node = (const float*)d_in[0];  // [8192, 2, 32] f32 -> rows of 64
    const float* edge = (const float*)d_in[1];  // [8192, 2, 32] f32 -> rows of 64
    float* out = (float*)d_out;                 // [8192, 8192] f32

    dim3 grid(NN / 128, NN / 128);  // 64 x 64 blocks
    dim3 block(256);                // 8 wave32 per block
    hipLaunchKernelGGL(ContextWeaver_wmma_f32, grid, block, 0, stream,
                       node, edge, out);
}